// ActorNew_64750926955165
// MI455X (gfx1250) — compile-verified
//
#include <hip/hip_runtime.h>
#include <math.h>

typedef __attribute__((ext_vector_type(2))) float v2f;
typedef __attribute__((ext_vector_type(8))) float v8f;

#define N_NODES 180224
#define N_EDGES 1441792
#define HIDDEN  128
#define NGRAPH  8192
#define FC_IN   (22 * HIDDEN)   // 2816

// ---------------- small elementwise kernels ----------------

__global__ void k_init_deg(float* __restrict__ deg) {
    int i = blockIdx.x * blockDim.x + threadIdx.x;
    if (i < N_NODES) deg[i] = 1.0f;                // self-loop weight
}

__global__ void k_deg_atomic(const long long* __restrict__ ei,
                             const float* __restrict__ ew,
                             float* __restrict__ deg) {
    int e = blockIdx.x * blockDim.x + threadIdx.x;
    if (e < N_EDGES) {
        long long col = ei[(long long)N_EDGES + e];
        atomicAdd(&deg[col], ew[e]);
    }
}

__global__ void k_dinv(float* __restrict__ deg) {
    int i = blockIdx.x * blockDim.x + threadIdx.x;
    if (i < N_NODES) {
        float d = deg[i];
        deg[i] = (d > 0.0f) ? rsqrtf(d) : 0.0f;
    }
}

// h = x @ W1   (K = 3, trivially memory-bound)
__global__ void k_lin1(const float* __restrict__ x,
                       const float* __restrict__ W1,
                       float* __restrict__ out) {
    long long idx = (long long)blockIdx.x * blockDim.x + threadIdx.x;
    if (idx >= (long long)N_NODES * HIDDEN) return;
    int i = (int)(idx >> 7);
    int j = (int)(idx & 127);
    const float* xr = x + (long long)i * 3;
    out[idx] = xr[0] * W1[j] + xr[1] * W1[HIDDEN + j] + xr[2] * W1[2 * HIDDEN + j];
}

// out[i][c] = dinv[i]^2 * in[i][c]   (self-loop contribution, also initializes out)
__global__ void k_selfinit(const float* __restrict__ in,
                           const float* __restrict__ dinv,
                           float* __restrict__ out) {
    long long idx = (long long)blockIdx.x * blockDim.x + threadIdx.x;
    if (idx >= (long long)N_NODES * HIDDEN) return;
    int i = (int)(idx >> 7);
    float d = dinv[i];
    out[idx] = d * d * in[idx];
}

// one wave32 per edge; each lane handles 4 channels (float4 gather, 4 f32 atomics scatter)
__global__ void k_agg(const float* __restrict__ h,
                      const long long* __restrict__ ei,
                      const float* __restrict__ ew,
                      const float* __restrict__ dinv,
                      float* __restrict__ out) {
    int e = (blockIdx.x * blockDim.x + threadIdx.x) >> 5;
    if (e >= N_EDGES) return;
    int lane = threadIdx.x & 31;
    long long row = ei[e];
    long long col = ei[(long long)N_EDGES + e];
    float norm = dinv[row] * ew[e] * dinv[col];
    const float4 v = ((const float4*)(h + row * HIDDEN))[lane];
    float* dst = out + col * HIDDEN + lane * 4;
    atomicAdd(dst + 0, norm * v.x);
    atomicAdd(dst + 1, norm * v.y);
    atomicAdd(dst + 2, norm * v.z);
    atomicAdd(dst + 3, norm * v.w);
}

__global__ void k_bias_act(float* __restrict__ buf,
                           const float* __restrict__ bias,
                           int act) {
    long long idx = (long long)blockIdx.x * blockDim.x + threadIdx.x;
    if (idx >= (long long)N_NODES * HIDDEN) return;
    int c = (int)(idx & 127);
    float v = buf[idx] + bias[c];
    if (act) v = (v > 0.0f) ? v : 0.01f * v;
    buf[idx] = v;
}

// ---------------- WMMA f32 GEMM:  C[M,128] = A[M,K] @ B[K,128] (+bias)(+lrelu) ----------
// Block = 256 threads = 8 waves; block handles one 16-row M tile; wave w handles cols [16w,16w+16).
// V_WMMA_F32_16X16X4_F32:  A 16x4 (lane=M, vgpr pair = K pair), B 4x16 (lane=N), C/D 8 vgprs.

__global__ void __launch_bounds__(256)
k_gemm_wmma(const float* __restrict__ A, const float* __restrict__ B,
            const float* __restrict__ bias, float* __restrict__ C,
            int M, int K, int act) {
    const int N = 128;
    int lane = threadIdx.x & 31;
    int wave = threadIdx.x >> 5;
    int m0 = blockIdx.x * 16;
    int n0 = wave * 16;
    int mr = lane & 15;            // A row / B col within tile
    int kh = (lane >> 4) << 1;     // 0 or 2: K sub-offset for this half-wave

    const float* arow = A + (long long)(m0 + mr) * K;
    const float* bcol = B + n0 + mr;

    v8f acc = {0.f, 0.f, 0.f, 0.f, 0.f, 0.f, 0.f, 0.f};

    for (int k0 = 0; k0 < K; k0 += 4) {
        v2f a, b;
        float2 av = *(const float2*)(arow + k0 + kh);   // contiguous pair -> b64 load
        a.x = av.x;
        a.y = av.y;
        b.x = bcol[(long long)(k0 + kh) * N];
        b.y = bcol[(long long)(k0 + kh + 1) * N];
        acc = __builtin_amdgcn_wmma_f32_16x16x4_f32(
            false, a, false, b, (short)0, acc, false, false);
    }

    int half = lane >> 4;          // 0 -> rows 0..7, 1 -> rows 8..15
    int cc = lane & 15;
    float bv = bias ? bias[n0 + cc] : 0.0f;
#pragma unroll
    for (int v = 0; v < 8; ++v) {
        int r = v + half * 8;
        float o = acc[v] + bv;
        if (act) o = (o > 0.0f) ? o : 0.01f * o;
        C[(long long)(m0 + r) * N + n0 + cc] = o;
    }
}

// ---------------- final head: one wave per graph, dot(z,Wout)+bout -> tanh*90+150 ------

__global__ void k_out(const float* __restrict__ z,
                      const float* __restrict__ Wout,
                      const float* __restrict__ bout,
                      float* __restrict__ out) {
    int g = (blockIdx.x * blockDim.x + threadIdx.x) >> 5;
    if (g >= NGRAPH) return;
    int lane = threadIdx.x & 31;
    float4 zv = ((const float4*)(z + (long long)g * HIDDEN))[lane];
    float4 wv = ((const float4*)Wout)[lane];
    float s = zv.x * wv.x + zv.y * wv.y + zv.z * wv.z + zv.w * wv.w;
#pragma unroll
    for (int off = 16; off > 0; off >>= 1)
        s += __shfl_down(s, off, 32);
    if (lane == 0)
        out[g] = tanhf(s + bout[0]) * 90.0f + 150.0f;
}

// -------------------------------- launcher ---------------------------------------------

extern "C" void kernel_launch(void* const* d_in, const int* in_sizes, int n_in,
                              void* d_out, int out_size, void* d_ws, size_t ws_size,
                              hipStream_t stream) {
    const float*     x    = (const float*)d_in[0];
    const long long* ei   = (const long long*)d_in[1];   // int64 [2,E]
    const float*     ew   = (const float*)d_in[2];
    const float*     W1   = (const float*)d_in[3];
    const float*     b1   = (const float*)d_in[4];
    const float*     W2   = (const float*)d_in[5];
    const float*     b2   = (const float*)d_in[6];
    const float*     Wf0  = (const float*)d_in[7];
    const float*     bf0  = (const float*)d_in[8];
    const float*     Wf1  = (const float*)d_in[9];
    const float*     bf1  = (const float*)d_in[10];
    const float*     Wout = (const float*)d_in[11];
    const float*     bout = (const float*)d_in[12];
    float*           out  = (float*)d_out;

    char* w = (char*)d_ws;
    size_t off = 0;
    float* dinv = (float*)(w + off); off += (size_t)N_NODES * 4;            // 720896 (256-mult)
    float* bufA = (float*)(w + off); off += (size_t)N_NODES * HIDDEN * 4;   // 92 MB
    float* bufB = (float*)(w + off); off += (size_t)N_NODES * HIDDEN * 4;   // 92 MB
    float* bufC = (float*)(w + off); off += (size_t)NGRAPH * HIDDEN * 4;    // 4 MB
    float* bufD = (float*)(w + off); off += (size_t)NGRAPH * HIDDEN * 4;    // 4 MB

    const int T = 256;
    long long nm = (long long)N_NODES * HIDDEN;
    int g_nm   = (int)((nm + T - 1) / T);
    int g_n    = (N_NODES + T - 1) / T;
    int g_e    = (N_EDGES + T - 1) / T;
    int g_agg  = (int)(((long long)N_EDGES * 32 + T - 1) / T);

    // degrees + inverse-sqrt norm
    k_init_deg  <<<g_n, T, 0, stream>>>(dinv);
    k_deg_atomic<<<g_e, T, 0, stream>>>(ei, ew, dinv);
    k_dinv      <<<g_n, T, 0, stream>>>(dinv);

    // GCN layer 1
    k_lin1      <<<g_nm, T, 0, stream>>>(x, W1, bufA);
    k_selfinit  <<<g_nm, T, 0, stream>>>(bufA, dinv, bufB);
    k_agg       <<<g_agg, T, 0, stream>>>(bufA, ei, ew, dinv, bufB);
    k_bias_act  <<<g_nm, T, 0, stream>>>(bufB, b1, 1);

    // GCN layer 2 (WMMA GEMM then aggregate)
    k_gemm_wmma <<<N_NODES / 16, T, 0, stream>>>(bufB, W2, nullptr, bufA, N_NODES, HIDDEN, 0);
    k_selfinit  <<<g_nm, T, 0, stream>>>(bufA, dinv, bufB);
    k_agg       <<<g_agg, T, 0, stream>>>(bufA, ei, ew, dinv, bufB);
    k_bias_act  <<<g_nm, T, 0, stream>>>(bufB, b2, 0);

    // MLP head (bufB viewed as [8192, 2816])
    k_gemm_wmma <<<NGRAPH / 16, T, 0, stream>>>(bufB, Wf0, bf0, bufC, NGRAPH, FC_IN, 1);
    k_gemm_wmma <<<NGRAPH / 16, T, 0, stream>>>(bufC, Wf1, bf1, bufD, NGRAPH, HIDDEN, 1);
    k_out       <<<NGRAPH * 32 / T, T, 0, stream>>>(bufD, Wout, bout, out);
}